// MolecularGNN_85650237817597
// MI455X (gfx1250) — compile-verified
//
#include <hip/hip_runtime.h>

#define NN 100000
#define NE 400000
#define NG 2000
#define KP1 80   // layer-1 K (78) padded to a multiple of 16

typedef float v2f __attribute__((ext_vector_type(2)));
typedef float v8f __attribute__((ext_vector_type(8)));

// ---------------------------------------------------------------- utilities
__global__ void set_val_kernel(float* __restrict__ p, int n, float v) {
    int i = blockIdx.x * blockDim.x + threadIdx.x;
    if (i < n) p[i] = v;
}

__global__ void deg_edges_kernel(const int* __restrict__ dst, float* __restrict__ deg, int E) {
    int i = blockIdx.x * blockDim.x + threadIdx.x;
    if (i < E) atomicAdd(&deg[dst[i]], 1.0f);
}

__global__ void rsqrt_kernel(float* __restrict__ p, int n) {
    int i = blockIdx.x * blockDim.x + threadIdx.x;
    if (i < n) p[i] = rsqrtf(p[i]);
}

// pad x: [n x 78] -> [n x 80], zero-filled tail columns
__global__ void pad_x_kernel(const float* __restrict__ x, float* __restrict__ xp, int n) {
    int i = blockIdx.x * blockDim.x + threadIdx.x;   // over n*80
    if (i < n * KP1) {
        int row = i / KP1;
        int col = i - row * KP1;
        xp[i] = (col < 78) ? x[row * 78 + col] : 0.f;
    }
}

// pad W1: [78 x 128] -> [80 x 128], zero rows 78..79
__global__ void pad_w1_kernel(const float* __restrict__ w, float* __restrict__ wp) {
    int i = blockIdx.x * blockDim.x + threadIdx.x;   // over 80*128
    if (i < KP1 * 128) wp[i] = (i < 78 * 128) ? w[i] : 0.f;
}

// ------------------------------------------------- fp32 WMMA GEMM: H = X @ W
// X: M x K row-major, W: K x F row-major, H: M x F row-major.
// Requires K % 16 == 0 and F % 64 == 0. MGUARD only when M % 32 != 0.
// Block tile 32x64, 8 waves, one 16x16 WMMA tile per wave, BK = 16.
#define BM 32
#define BN 64
#define BK 16

template <bool MGUARD>
__global__ __launch_bounds__(256)
void gemm_wmma_kernel(const float* __restrict__ X, const float* __restrict__ W,
                      float* __restrict__ H, int M, int K, int F) {
    __shared__ float As[BM][20];   // pad 20: conflict-free column reads
    __shared__ float Bs[BK][68];   // pad 68: halves land on distinct banks

    const int t    = threadIdx.x;
    const int wave = t >> 5;
    const int lane = t & 31;
    const int half = lane >> 4;      // 0 | 1
    const int lo   = lane & 15;
    const int trow = (wave & 1) << 4;   // 0 | 16
    const int tcol = (wave >> 1) << 4;  // 0..48
    const int m0   = blockIdx.x * BM;
    const int n0   = blockIdx.y * BN;

    // A-tile mapping: all 256 threads, 2 floats each (32 rows x 16 cols)
    const int ar = t >> 3;           // 0..31
    const int ac = (t & 7) << 1;     // 0,2,..,14
    // B-tile mapping: all 256 threads, 4 floats each (16 rows x 64 cols)
    const int br = t >> 4;           // 0..15
    const int bc = (t & 15) << 2;    // 0,4,..,60

    v8f acc = {0.f, 0.f, 0.f, 0.f, 0.f, 0.f, 0.f, 0.f};

    for (int kb = 0; kb < K; kb += BK) {
        // stage A tile (32 x 16): one b64 per thread, no divergence
        {
            float2 v = {0.f, 0.f};
            if (!MGUARD || (m0 + ar) < M) {
                const float* p = X + (long long)(m0 + ar) * K + kb + ac;
                v = *(const float2*)p;
                if ((t & 7) == 0 && kb + BK < K)
                    __builtin_prefetch(p + BK, 0, 1);   // global_prefetch_b8
            }
            As[ar][ac + 0] = v.x;
            As[ar][ac + 1] = v.y;
        }
        // stage B tile (16 x 64): one b128 per thread
        {
            float4 v = *(const float4*)(W + (long long)(kb + br) * F + n0 + bc);
            Bs[br][bc + 0] = v.x; Bs[br][bc + 1] = v.y;
            Bs[br][bc + 2] = v.z; Bs[br][bc + 3] = v.w;
        }
        __syncthreads();

#pragma unroll
        for (int kk = 0; kk < BK; kk += 4) {
            int ka = kk + (half << 1);
            v2f a, b;
            a.x = As[trow + lo][ka + 0];
            a.y = As[trow + lo][ka + 1];
            b.x = Bs[ka + 0][tcol + lo];
            b.y = Bs[ka + 1][tcol + lo];
            acc = __builtin_amdgcn_wmma_f32_16x16x4_f32(
                false, a, false, b, (short)0, acc, false, false);
        }
        __syncthreads();
    }

    // c[r] holds row (trow + r + half*8), col (tcol + lo)
#pragma unroll
    for (int r = 0; r < 8; ++r) {
        int grow = m0 + trow + (half << 3) + r;
        if (!MGUARD || grow < M) H[(long long)grow * F + n0 + tcol + lo] = acc[r];
    }
}

// ------------------------------------------ GCN aggregation (symmetric norm)
// out[i,:] = h[i,:] * dinv[i]^2   (self-loop term; also initializes out)
__global__ void selfloop_kernel(const float* __restrict__ h, const float* __restrict__ dinv,
                                float* __restrict__ out, int total, int lgF) {
    int i = blockIdx.x * blockDim.x + threadIdx.x;
    if (i < total) {
        float di = dinv[i >> lgF];
        out[i] = h[i] * di * di;
    }
}

// out[dst,:] += h[src,:] * dinv[src]*dinv[dst] for each edge; float4 chunks.
__global__ void edge_agg_kernel(const float* __restrict__ h, const int* __restrict__ src,
                                const int* __restrict__ dst, const float* __restrict__ dinv,
                                float* __restrict__ out, int E, int F, int lgC) {
    long long tid = (long long)blockIdx.x * blockDim.x + threadIdx.x;
    long long e = tid >> lgC;
    int fc = (int)(tid & ((1 << lgC) - 1));
    if (e >= E) return;
    int s = src[e], d = dst[e];
    float c = dinv[s] * dinv[d];
    float4 hv = *(const float4*)(h + (long long)s * F + (fc << 2));
    float* op = out + (long long)d * F + (fc << 2);
    atomicAdd(op + 0, hv.x * c);
    atomicAdd(op + 1, hv.y * c);
    atomicAdd(op + 2, hv.z * c);
    atomicAdd(op + 3, hv.w * c);
}

// in-place x = relu(x + bias[f]);  F is a power of two
__global__ void bias_relu_kernel(float* __restrict__ x, const float* __restrict__ b,
                                 int total, int F) {
    int i = blockIdx.x * blockDim.x + threadIdx.x;
    if (i < total) x[i] = fmaxf(x[i] + b[i & (F - 1)], 0.f);
}

// ------------------------------------------------------------- mean pooling
__global__ void pool_cnt_kernel(const int* __restrict__ batch, float* __restrict__ cnts, int n) {
    int i = blockIdx.x * blockDim.x + threadIdx.x;
    if (i < n) atomicAdd(&cnts[batch[i]], 1.0f);
}

__global__ void pool_sum_kernel(const float* __restrict__ h, const int* __restrict__ batch,
                                float* __restrict__ sums, int n) {  // F = 128
    long long tid = (long long)blockIdx.x * blockDim.x + threadIdx.x;
    long long i = tid >> 5;           // 32 float4 chunks per node
    int fc = (int)(tid & 31);
    if (i >= n) return;
    int g = batch[i];
    float4 hv = *(const float4*)(h + i * 128 + (fc << 2));
    float* op = sums + (long long)g * 128 + (fc << 2);
    atomicAdd(op + 0, hv.x);
    atomicAdd(op + 1, hv.y);
    atomicAdd(op + 2, hv.z);
    atomicAdd(op + 3, hv.w);
}

__global__ void pool_div_kernel(const float* __restrict__ sums, const float* __restrict__ cnts,
                                float* __restrict__ pooled, int total) {  // F = 128
    int i = blockIdx.x * blockDim.x + threadIdx.x;
    if (i < total) pooled[i] = sums[i] / fmaxf(cnts[i >> 7], 1.0f);
}

// ------------------------------------------------------------------ head MLP
__global__ void fc2_kernel(const float* __restrict__ h, const float* __restrict__ w,
                           const float* __restrict__ b, float* __restrict__ out, int G) {
    int g = blockIdx.x * blockDim.x + threadIdx.x;
    if (g >= G) return;
    float acc = b[0];
    const float* hp = h + (long long)g * 64;
#pragma unroll 8
    for (int j = 0; j < 64; ++j) acc += hp[j] * w[j];
    out[g] = acc;
}

// ------------------------------------------------------------------- driver
static inline int cdiv(long long a, int b) { return (int)((a + b - 1) / b); }

extern "C" void kernel_launch(void* const* d_in, const int* in_sizes, int n_in,
                              void* d_out, int out_size, void* d_ws, size_t ws_size,
                              hipStream_t stream) {
    const float* x   = (const float*)d_in[0];
    const int*   ei  = (const int*)d_in[1];
    const int*   src = ei;
    const int*   dst = ei + NE;
    const int*   batch = (const int*)d_in[2];
    const float* W1 = (const float*)d_in[3];  const float* b1 = (const float*)d_in[4];
    const float* W2 = (const float*)d_in[5];  const float* b2 = (const float*)d_in[6];
    const float* W3 = (const float*)d_in[7];  const float* b3 = (const float*)d_in[8];
    const float* fw1 = (const float*)d_in[9]; const float* fb1 = (const float*)d_in[10];
    const float* fw2 = (const float*)d_in[11];const float* fb2 = (const float*)d_in[12];
    float* out = (float*)d_out;

    // workspace layout (floats)
    float* Abuf   = (float*)d_ws;                    // N x 256  (GEMM output h)
    float* Bbuf   = Abuf + (size_t)NN * 256;         // N x 256  (agg / activation)
    float* Xp     = Bbuf + (size_t)NN * 256;         // N x 80   (padded input)
    float* W1p    = Xp + (size_t)NN * KP1;           // 80 x 128 (padded W1)
    float* dinv   = W1p + (size_t)KP1 * 128;         // N        (deg -> rsqrt)
    float* sums   = dinv + NN;                       // G x 128
    float* cnts   = sums + (size_t)NG * 128;         // G
    float* pooled = cnts + NG;                       // G x 128
    float* fc1h   = pooled + (size_t)NG * 128;       // G x 64

    const int T = 256;

    // degrees with self-loops -> dinv
    set_val_kernel<<<cdiv(NN, T), T, 0, stream>>>(dinv, NN, 1.0f);
    deg_edges_kernel<<<cdiv(NE, T), T, 0, stream>>>(dst, dinv, NE);
    rsqrt_kernel<<<cdiv(NN, T), T, 0, stream>>>(dinv, NN);

    // pad layer-1 operands so all GEMMs have K % 16 == 0
    pad_x_kernel<<<cdiv((long long)NN * KP1, T), T, 0, stream>>>(x, Xp, NN);
    pad_w1_kernel<<<cdiv(KP1 * 128, T), T, 0, stream>>>(W1, W1p);

    // ---- layer 1: 80 -> 128
    gemm_wmma_kernel<false><<<dim3(cdiv(NN, BM), 128 / BN), T, 0, stream>>>(Xp, W1p, Abuf, NN, KP1, 128);
    selfloop_kernel<<<cdiv((long long)NN * 128, T), T, 0, stream>>>(Abuf, dinv, Bbuf, NN * 128, 7);
    edge_agg_kernel<<<cdiv((long long)NE * 32, T), T, 0, stream>>>(Abuf, src, dst, dinv, Bbuf, NE, 128, 5);
    bias_relu_kernel<<<cdiv((long long)NN * 128, T), T, 0, stream>>>(Bbuf, b1, NN * 128, 128);

    // ---- layer 2: 128 -> 256
    gemm_wmma_kernel<false><<<dim3(cdiv(NN, BM), 256 / BN), T, 0, stream>>>(Bbuf, W2, Abuf, NN, 128, 256);
    selfloop_kernel<<<cdiv((long long)NN * 256, T), T, 0, stream>>>(Abuf, dinv, Bbuf, NN * 256, 8);
    edge_agg_kernel<<<cdiv((long long)NE * 64, T), T, 0, stream>>>(Abuf, src, dst, dinv, Bbuf, NE, 256, 6);
    bias_relu_kernel<<<cdiv((long long)NN * 256, T), T, 0, stream>>>(Bbuf, b2, NN * 256, 256);

    // ---- layer 3: 256 -> 128
    gemm_wmma_kernel<false><<<dim3(cdiv(NN, BM), 128 / BN), T, 0, stream>>>(Bbuf, W3, Abuf, NN, 256, 128);
    selfloop_kernel<<<cdiv((long long)NN * 128, T), T, 0, stream>>>(Abuf, dinv, Bbuf, NN * 128, 7);
    edge_agg_kernel<<<cdiv((long long)NE * 32, T), T, 0, stream>>>(Abuf, src, dst, dinv, Bbuf, NE, 128, 5);
    bias_relu_kernel<<<cdiv((long long)NN * 128, T), T, 0, stream>>>(Bbuf, b3, NN * 128, 128);

    // ---- global mean pool (sums and cnts are contiguous -> one zero pass)
    set_val_kernel<<<cdiv(NG * 128 + NG, T), T, 0, stream>>>(sums, NG * 128 + NG, 0.0f);
    pool_cnt_kernel<<<cdiv(NN, T), T, 0, stream>>>(batch, cnts, NN);
    pool_sum_kernel<<<cdiv((long long)NN * 32, T), T, 0, stream>>>(Bbuf, batch, sums, NN);
    pool_div_kernel<<<cdiv(NG * 128, T), T, 0, stream>>>(sums, cnts, pooled, NG * 128);

    // ---- head: fc1 (WMMA, M=2000 needs the M guard) + relu, fc2
    gemm_wmma_kernel<true><<<dim3(cdiv(NG, BM), 64 / BN), T, 0, stream>>>(pooled, fw1, fc1h, NG, 128, 64);
    bias_relu_kernel<<<cdiv(NG * 64, T), T, 0, stream>>>(fc1h, fb1, NG * 64, 64);
    fc2_kernel<<<cdiv(NG, T), T, 0, stream>>>(fc1h, fw2, fb2, out, NG);
}